// GraphAttentionLayer_30769145708916
// MI455X (gfx1250) — compile-verified
//
#include <hip/hip_runtime.h>
#include <hip/hip_bf16.h>
#include <math.h>

// GAT layer: B=8, N=2048, F_in=128, F_out=64
#define GB    8
#define GN    2048
#define GFIN  128
#define GFOUT 64
#define GSLOPE 0.2f
#define GNEGBIG (-9.0e15f)

typedef __attribute__((ext_vector_type(2))) float v2f;
typedef __attribute__((ext_vector_type(8))) float v8f;

// ---------------------------------------------------------------------------
// Kernel A: h = input @ W  (16 n-rows per workgroup, fp32 WMMA 16x16x4),
//           s1 = h @ a1, s2 = h @ a2
// ---------------------------------------------------------------------------
__global__ __launch_bounds__(128) void gat_h_kernel(
    const float* __restrict__ inp,   // (B,N,128)
    const float* __restrict__ W,     // (128,64)
    const float* __restrict__ av,    // (128,) : a1=av[0:64], a2=av[64:128]
    float* __restrict__ h,           // ws (B,N,64)
    float* __restrict__ s1,          // ws (B,N)
    float* __restrict__ s2)          // ws (B,N)
{
    const int b  = blockIdx.x >> 7;          // N/16 = 128 tiles per batch
    const int n0 = (blockIdx.x & 127) << 4;
    const int tid  = threadIdx.x;            // 0..127 (4 waves)
    const int lane = tid & 31;
    const int wave = tid >> 5;

    __shared__ float sInp[16][GFIN];         // 8 KB
    __shared__ float sW[GFIN][GFOUT];        // 32 KB
    __shared__ float sH[16][GFOUT];          // 4 KB

    for (int e = tid; e < 16 * GFIN; e += 128) {
        int r = e >> 7, c = e & 127;
        sInp[r][c] = inp[((size_t)b * GN + n0 + r) * GFIN + c];
    }
    for (int e = tid; e < GFIN * GFOUT; e += 128) {
        sW[e >> 6][e & 63] = W[e];
    }
    __syncthreads();

    const int o_base  = wave << 4;           // 4 waves x 16 output cols = 64
    const int halfsel = lane >> 4;           // 0: K={0,1}, 1: K={2,3}
    const int lrow    = lane & 15;

    v8f c = {};
    for (int k = 0; k < GFIN; k += 4) {
        const int ka = k + (halfsel << 1);
        v2f a, bb;
        a.x  = sInp[lrow][ka + 0];
        a.y  = sInp[lrow][ka + 1];
        bb.x = sW[ka + 0][o_base + lrow];
        bb.y = sW[ka + 1][o_base + lrow];
        c = __builtin_amdgcn_wmma_f32_16x16x4_f32(false, a, false, bb,
                                                  (short)0, c, false, false);
    }

    // C layout: vgpr r, lanes 0-15 -> row r; lanes 16-31 -> row r+8
    for (int r = 0; r < 8; ++r) {
        const int row = r + (halfsel << 3);
        const float v = c[r];
        sH[row][o_base + lrow] = v;
        h[((size_t)b * GN + n0 + row) * GFOUT + o_base + lrow] = v;
    }
    __syncthreads();

    if (tid < 16) {
        float acc1 = 0.f, acc2 = 0.f;
        for (int o = 0; o < GFOUT; ++o) {
            const float hv = sH[tid][o];
            acc1 = fmaf(hv, av[o], acc1);
            acc2 = fmaf(hv, av[GFOUT + o], acc2);
        }
        s1[(size_t)b * GN + n0 + tid] = acc1;
        s2[(size_t)b * GN + n0 + tid] = acc2;
    }
}

// ---------------------------------------------------------------------------
// Kernel B: masked leaky-relu scores -> softmax -> attention (non-temporal
// streamed out) -> h_prime = attention @ h via fp32 WMMA -> relu -> out.
// One workgroup per (b, 16 i-rows); j processed in 64-wide tiles whose h
// panel is DMA'd to LDS with GLOBAL_LOAD_ASYNC_TO_LDS_B128 (ASYNCcnt),
// overlapped with the softmax/attention-store VALU work.
// ---------------------------------------------------------------------------
__global__ __launch_bounds__(128) void gat_attn_kernel(
    const int*   __restrict__ adj,   // (N,N)
    const float* __restrict__ h,     // (B,N,64)
    const float* __restrict__ s1,    // (B,N)
    const float* __restrict__ s2,    // (B,N)
    float* __restrict__ out,         // (B,N,64)
    float* __restrict__ attn_out)    // (B,N,N)
{
    const int b  = blockIdx.x >> 7;
    const int i0 = (blockIdx.x & 127) << 4;
    const int tid  = threadIdx.x;            // 0..127
    const int lane = tid & 31;
    const int wave = tid >> 5;

    __shared__ float sS2[GN];                // 8 KB: s2 for the whole batch row
    __shared__ float sS1[16];
    __shared__ float sRowMax[16];
    __shared__ float sRowInvZ[16];
    __shared__ float sRedM[16][9];
    __shared__ float sRedS[16][9];
    __shared__ float sP[16][64];             // 4 KB  attention tile (A operand)
    __shared__ float sH[64][GFOUT];          // 16 KB h panel        (B operand)

    for (int e = tid; e < GN; e += 128)
        sS2[e] = s2[(size_t)b * GN + e];
    if (tid < 16)
        sS1[tid] = s1[(size_t)b * GN + i0 + tid];
    __syncthreads();

    // ---- pass 1: online (max, sum-of-exp) per row; 8 threads per row ----
    {
        const int row = tid >> 3;            // 0..15
        const int g   = tid & 7;             // 0..7
        const float se1 = sS1[row];
        const size_t arow = (size_t)(i0 + row) * GN;
        float m = -INFINITY, ssum = 0.f;
        for (int j = g; j < GN; j += 8) {
            float e = se1 + sS2[j];
            e = (e > 0.f) ? e : GSLOPE * e;
            const float x = (adj[arow + j] > 0) ? e : GNEGBIG;
            const float mn = fmaxf(m, x);
            ssum = ssum * __expf(m - mn) + __expf(x - mn);
            m = mn;
        }
        sRedM[row][g] = m;
        sRedS[row][g] = ssum;
    }
    __syncthreads();
    if (tid < 16) {
        float m = -INFINITY, s = 0.f;
        for (int g = 0; g < 8; ++g) {
            const float mg = sRedM[tid][g];
            const float mn = fmaxf(m, mg);
            s = s * __expf(m - mn) + sRedS[tid][g] * __expf(mg - mn);
            m = mn;
        }
        sRowMax[tid]  = m;
        sRowInvZ[tid] = 1.f / s;
    }

    // ---- pass 2: attention tiles + WMMA accumulation of h_prime ----
    const int o_base  = wave << 4;
    const int halfsel = lane >> 4;
    const int lrow    = lane & 15;
    const unsigned int sH0 = (unsigned int)(uintptr_t)(&sH[0][0]); // LDS byte addr
    const unsigned long long hB =
        (unsigned long long)(uintptr_t)(h + (size_t)b * GN * GFOUT);
    v8f acc = {};

    for (int jt = 0; jt < GN; jt += 64) {
        __syncthreads();  // previous sP/sH fully consumed (also covers pass 1)

        // async DMA: 64x64 f32 h panel -> LDS (1024 x b128, 8 per thread)
        for (int it = 0; it < 8; ++it) {
            const int q  = (it << 7) + tid;          // b128 chunk id 0..1023
            const int r  = q >> 4;                   // 64 rows
            const int c4 = q & 15;                   // 16 x 16B per row
            const unsigned int goff = (unsigned int)(((jt + r) << 8) + (c4 << 4));
            const unsigned int loff = sH0 + (unsigned int)((r << 8) + (c4 << 4));
            asm volatile("global_load_async_to_lds_b128 %0, %1, %2"
                         :: "v"(loff), "v"(goff), "s"(hB) : "memory");
        }

        // attention probabilities while the DMA is in flight:
        // P tile (16 x 64), 8 per thread; rows stream out as 256B
        // non-temporal bursts (the dominant 134 MB of HBM traffic)
        for (int e = tid; e < 16 * 64; e += 128) {
            const int r  = e >> 6, cc = e & 63;
            const int j  = jt + cc;
            float ev = sS1[r] + sS2[j];
            ev = (ev > 0.f) ? ev : GSLOPE * ev;
            const float x = (adj[(size_t)(i0 + r) * GN + j] > 0) ? ev : GNEGBIG;
            const float p = __expf(x - sRowMax[r]) * sRowInvZ[r];
            sP[r][cc] = p;
            __builtin_nontemporal_store(p, &attn_out[((size_t)b * GN + i0 + r) * GN + j]);
        }

        asm volatile("s_wait_asynccnt 0x0" ::: "memory");
        __syncthreads();

        // acc(16x16) += P(16x64) x H(64x16-slice), 16 K-steps of 4
        for (int k = 0; k < 64; k += 4) {
            const int ka = k + (halfsel << 1);
            v2f a, bb;
            a.x  = sP[lrow][ka + 0];
            a.y  = sP[lrow][ka + 1];
            bb.x = sH[ka + 0][o_base + lrow];
            bb.y = sH[ka + 1][o_base + lrow];
            acc = __builtin_amdgcn_wmma_f32_16x16x4_f32(false, a, false, bb,
                                                        (short)0, acc, false, false);
        }
    }

    // relu + store h_prime
    for (int r = 0; r < 8; ++r) {
        const int row = r + (halfsel << 3);
        out[((size_t)b * GN + i0 + row) * GFOUT + o_base + lrow] = fmaxf(acc[r], 0.f);
    }
}

// ---------------------------------------------------------------------------
extern "C" void kernel_launch(void* const* d_in, const int* in_sizes, int n_in,
                              void* d_out, int out_size, void* d_ws, size_t ws_size,
                              hipStream_t stream) {
    const float* inp = (const float*)d_in[0];   // (8,2048,128)
    const int*   adj = (const int*)  d_in[1];   // (2048,2048)
    const float* W   = (const float*)d_in[2];   // (128,64)
    const float* av  = (const float*)d_in[3];   // (128,1)

    float* out_hp = (float*)d_out;                                  // (8,2048,64)
    float* out_at = (float*)d_out + (size_t)GB * GN * GFOUT;        // (8,2048,2048,1)

    char* ws = (char*)d_ws;
    float* h  = (float*)ws;                                   // 4 MB
    float* s1 = (float*)(ws + (size_t)GB * GN * GFOUT * 4);   // 64 KB
    float* s2 = (float*)(ws + (size_t)GB * GN * GFOUT * 4 + (size_t)GB * GN * 4);

    const int nblocks = GB * (GN / 16);  // 1024
    gat_h_kernel<<<nblocks, 128, 0, stream>>>(inp, W, av, h, s1, s2);
    gat_attn_kernel<<<nblocks, 128, 0, stream>>>(adj, h, s1, s2, out_hp, out_at);
}